// SelfAttention_37632503448228
// MI455X (gfx1250) — compile-verified
//
#include <hip/hip_runtime.h>
#include <math.h>

typedef __attribute__((ext_vector_type(2))) float        v2f;
typedef __attribute__((ext_vector_type(8))) float        v8f;
typedef __attribute__((ext_vector_type(4))) unsigned int u32x4;
typedef __attribute__((ext_vector_type(8))) int          i32x8;
typedef __attribute__((ext_vector_type(4))) int          i32x4;

#define HW_N   1024
#define C_IN   96
#define OC_QKV 288
#define NBATCH 4
#define SCALE_F 0.17677669529663687f   // 32^-0.5

// ---------- reductions within a 16-lane half of a wave32 ----------
__device__ __forceinline__ float hmax16(float v) {
    v = fmaxf(v, __shfl_xor(v, 1, 16));
    v = fmaxf(v, __shfl_xor(v, 2, 16));
    v = fmaxf(v, __shfl_xor(v, 4, 16));
    v = fmaxf(v, __shfl_xor(v, 8, 16));
    return v;
}
__device__ __forceinline__ float hsum16(float v) {
    v += __shfl_xor(v, 1, 16);
    v += __shfl_xor(v, 2, 16);
    v += __shfl_xor(v, 4, 16);
    v += __shfl_xor(v, 8, 16);
    return v;
}

// ---------------------------------------------------------------------------
// TDM: DMA a 2D tile (3 rows x 1024 f32, row stride 96*1024 f32) from global
// memory into LDS (planar, row-major). Descriptor per CDNA5 ISA.
//   group0: count=1 | lds_addr[63:32] | global_addr[120:64] | type=2
//   group1: data_size=4B, tensor_dim0=1024, tensor_dim1=3,
//           tile_dim0=1024, tile_dim1=3, tensor_dim0_stride=98304
//   groups 2/3: zero (2D tile, tile_dim2 unused)
// Issued by one wave; completion via TENSORcnt.
// ---------------------------------------------------------------------------
__device__ __forceinline__ void tdm_load_3xHW(const float* gsrc, void* ldsDst)
{
    const unsigned long long ga = (unsigned long long)(uintptr_t)gsrc;
    u32x4 g0;
    g0.x = 1u;                                          // count = 1 (valid D#)
    g0.y = (unsigned)(uintptr_t)ldsDst;                 // LDS byte address
    g0.z = (unsigned)(ga & 0xFFFFFFFFu);                // global_addr[31:0]
    g0.w = (unsigned)((ga >> 32) & 0x1FFFFFFu)          // global_addr[56:32]
         | (2u << 30);                                  // type = 2 ("image")
    const i32x8 g1 = { 0x00020000,                      // data_size=2 (4B)
                       (int)(1024u << 16),              // tensor_dim0 = 1024
                       (int)(3u    << 16),              // tensor_dim1 = 3
                       (int)(1024u << 16),              // tile_dim0   = 1024
                       3,                               // tile_dim1   = 3
                       98304,                           // dim0_stride = 96*1024
                       0, 0 };
    const i32x4 z4 = { 0, 0, 0, 0 };
#if __clang_major__ >= 23
    const i32x8 z8 = { 0, 0, 0, 0, 0, 0, 0, 0 };
    __builtin_amdgcn_tensor_load_to_lds(g0, g1, z4, z4, z8, 0);
#else
    __builtin_amdgcn_tensor_load_to_lds(g0, g1, z4, z4, 0);
#endif
}

// ---------------------------------------------------------------------------
// Generic per-batch channel GEMM with fp32 WMMA (16x16x4):
//   dst[b][o][n] = sum_c W[o][c] * src[b][c][n] + bias[o],   n in [0,1024)
// One wave per 16x16 output tile; K loop chains WMMAs over C in steps of 4.
// ---------------------------------------------------------------------------
__global__ __launch_bounds__(256) void gemm16_wmma_f32(
    const float* __restrict__ W, const float* __restrict__ bias,
    const float* __restrict__ src, float* __restrict__ dst, int O, int C)
{
    const int lane = threadIdx.x & 31;
    const int wave = threadIdx.x >> 5;
    const int tile = blockIdx.x * 8 + wave;
    const int tilesPerBatch = (O >> 4) * (HW_N >> 4);
    const int b  = tile / tilesPerBatch;
    const int t  = tile % tilesPerBatch;
    const int ot = t / (HW_N >> 4);
    const int nt = t % (HW_N >> 4);
    const int half = lane >> 4;          // 0: K=0..1, 1: K=2..3
    const int l15  = lane & 15;
    const int rowA = ot * 16 + l15;      // A row (output channel)
    const int colB = nt * 16 + l15;      // B column (spatial)
    const float* srcB = src + (size_t)b * C * HW_N;

    v8f acc = {};
    for (int kb = 0; kb < C; kb += 4) {
        const int kk = kb + half * 2;
        v2f a, bf;
        a.x  = W[rowA * C + kk];
        a.y  = W[rowA * C + kk + 1];
        bf.x = srcB[(size_t)kk * HW_N + colB];
        bf.y = srcB[(size_t)(kk + 1) * HW_N + colB];
        acc = __builtin_amdgcn_wmma_f32_16x16x4_f32(
                  false, a, false, bf, (short)0, acc, false, false);
    }

    float* dstB = dst + (size_t)b * O * HW_N;
#pragma unroll
    for (int r = 0; r < 8; ++r) {
        const int m = ot * 16 + r + half * 8;
        dstB[(size_t)m * HW_N + colB] = acc[r] + bias[m];
    }
}

// ---------------------------------------------------------------------------
// Fused attention for the reference's head-sliced q/k/v (contraction K = 3).
// One block per (b, d) pair (128 blocks, 8 waves).
//   * K and V tiles (3 x 1024 f32 each) DMA'd global->LDS by the Tensor Data
//     Mover (wave 0 issues, s_wait_tensorcnt, then block barrier).
//   * One-time repack planar -> interleaved float4 rows (zero-padded 4th
//     slot) so the hot loop uses ds_load_b64 (K^T frag) + ds_load_b128 (V).
//   * Per 16-row tile, two passes of 64 fp32 WMMA 16x16x4 tiles:
//       pass A: per-lane running row max, one shuffle reduction at the end;
//       pass B: p = exp(s - m), per-lane partial row-sum + partial PV,
//               cross-lane sums once at the end.
// Output scattered through the reference reshape: y[b][(n*96 + d*3 + j)].
// ---------------------------------------------------------------------------
__global__ __launch_bounds__(256) void attn_flash_wmma(
    const float* __restrict__ qkv, float* __restrict__ y)
{
    const int b = blockIdx.x >> 5;
    const int d = blockIdx.x & 31;

    __shared__ float  Kst[3][HW_N];     // TDM landing zone (planar)
    __shared__ float  Vst[3][HW_N];
    __shared__ float4 Klds[HW_N];       // interleaved, zero-padded
    __shared__ float4 Vlds[HW_N];

    const float* base = qkv + (size_t)b * OC_QKV * HW_N;
    // head slices: q = head0 (chans d,96+d,192+d), k = head1 (+32), v = head2 (+64)
    const float* kbase = base + (size_t)(32 + d) * HW_N;
    const float* vbase = base + (size_t)(64 + d) * HW_N;

    const int lane = threadIdx.x & 31;
    const int wave = threadIdx.x >> 5;

    if (wave == 0) {
        tdm_load_3xHW(kbase, &Kst[0][0]);
        tdm_load_3xHW(vbase, &Vst[0][0]);
        __builtin_amdgcn_s_wait_tensorcnt(0);
    }
    __syncthreads();

    for (int m = threadIdx.x; m < HW_N; m += 256) {
        Klds[m] = make_float4(Kst[0][m], Kst[1][m], Kst[2][m], 0.f);
        Vlds[m] = make_float4(Vst[0][m], Vst[1][m], Vst[2][m], 0.f);
    }
    __syncthreads();

    const int half = lane >> 4;
    const int l15  = lane & 15;

    const float* qc0 = base + (size_t)(  0 + d) * HW_N;
    const float* qc1 = base + (size_t)( 96 + d) * HW_N;
    const float* qc2 = base + (size_t)(192 + d) * HW_N;

    for (int rt = wave; rt < 64; rt += 8) {          // 16-row tiles
        const int n = rt * 16 + l15;
        v2f afrag;                                   // scale folded into Q
        if (half == 0) { afrag.x = SCALE_F * qc0[n]; afrag.y = SCALE_F * qc1[n]; }
        else           { afrag.x = SCALE_F * qc2[n]; afrag.y = 0.f; }

        // ---------------- pass A: row maxima ----------------
        float mrow[8];
#pragma unroll
        for (int r = 0; r < 8; ++r) mrow[r] = -3.402823466e38f;

        for (int ct = 0; ct < 64; ++ct) {
            const int mcol = ct * 16 + l15;
            const float* kr = (const float*)&Klds[mcol];
            v2f bfrag;
            bfrag.x = kr[half * 2];
            bfrag.y = kr[half * 2 + 1];
            v8f s = {};
            s = __builtin_amdgcn_wmma_f32_16x16x4_f32(
                    false, afrag, false, bfrag, (short)0, s, false, false);
#pragma unroll
            for (int r = 0; r < 8; ++r) mrow[r] = fmaxf(mrow[r], s[r]);
        }
#pragma unroll
        for (int r = 0; r < 8; ++r) mrow[r] = hmax16(mrow[r]);

        // -------- pass B: exp, partial row-sums, partial PV --------
        float lsum[8], acc0[8], acc1[8], acc2[8];
#pragma unroll
        for (int r = 0; r < 8; ++r) { lsum[r] = 0.f; acc0[r] = 0.f; acc1[r] = 0.f; acc2[r] = 0.f; }

        for (int ct = 0; ct < 64; ++ct) {
            const int mcol = ct * 16 + l15;
            const float* kr = (const float*)&Klds[mcol];
            v2f bfrag;
            bfrag.x = kr[half * 2];
            bfrag.y = kr[half * 2 + 1];
            v8f s = {};
            s = __builtin_amdgcn_wmma_f32_16x16x4_f32(
                    false, afrag, false, bfrag, (short)0, s, false, false);

            const float4 vv = Vlds[mcol];            // ds_load_b128
#pragma unroll
            for (int r = 0; r < 8; ++r) {
                const float p = __expf(s[r] - mrow[r]);
                lsum[r] += p;
                acc0[r] += p * vv.x;
                acc1[r] += p * vv.y;
                acc2[r] += p * vv.z;
            }
        }

        // final cross-lane reductions (once per row tile)
#pragma unroll
        for (int r = 0; r < 8; ++r) {
            lsum[r] = hsum16(lsum[r]);
            acc0[r] = hsum16(acc0[r]);
            acc1[r] = hsum16(acc1[r]);
            acc2[r] = hsum16(acc2[r]);
        }
        if (l15 < 3) {
            const int j = l15;
#pragma unroll
            for (int r = 0; r < 8; ++r) {
                const int   row = rt * 16 + r + half * 8;
                const float val = (j == 0 ? acc0[r] : (j == 1 ? acc1[r] : acc2[r]))
                                  / lsum[r];
                const size_t idx = (size_t)row * 96 + (size_t)d * 3 + j;
                y[(size_t)b * (96 * HW_N) + idx] = val;      // scrambled reshape
            }
        }
    }
}

extern "C" void kernel_launch(void* const* d_in, const int* in_sizes, int n_in,
                              void* d_out, int out_size, void* d_ws, size_t ws_size,
                              hipStream_t stream)
{
    (void)in_sizes; (void)n_in; (void)out_size; (void)ws_size;
    const float* x      = (const float*)d_in[0];
    const float* qkv_w  = (const float*)d_in[1];
    const float* qkv_b  = (const float*)d_in[2];
    const float* proj_w = (const float*)d_in[3];
    const float* proj_b = (const float*)d_in[4];
    float*       out    = (float*)d_out;

    float* qkv_ws = (float*)d_ws;                                  // 4*288*1024 f32
    float* y_ws   = qkv_ws + (size_t)NBATCH * OC_QKV * HW_N;       // 4*96*1024 f32

    // 1) QKV: 4 * 18 * 64 = 4608 tiles / 8 waves = 576 blocks
    gemm16_wmma_f32<<<576, 256, 0, stream>>>(qkv_w, qkv_b, x, qkv_ws, OC_QKV, C_IN);
    // 2) fused attention: one block per (b, head_dim) pair
    attn_flash_wmma<<<NBATCH * 32, 256, 0, stream>>>(qkv_ws, y_ws);
    // 3) proj: 4 * 6 * 64 = 1536 tiles / 8 = 192 blocks
    gemm16_wmma_f32<<<192, 256, 0, stream>>>(proj_w, proj_b, y_ws, out, C_IN, C_IN);
}